// L_LMCV_51290499449178
// MI455X (gfx1250) — compile-verified
//
#include <hip/hip_runtime.h>
#include <hip/hip_bf16.h>
#include <cstdint>

#define BATCH 16
#define CHN   3
#define HH    512
#define WW    512
#define KWIN  64
#define PL    31
#define PR    32
#define NPIX  ((size_t)BATCH * HH * WW)   // 4,194,304 pooled pixels

typedef unsigned int u32x4 __attribute__((ext_vector_type(4)));
typedef unsigned int u32x8 __attribute__((ext_vector_type(8)));
typedef float        v2f   __attribute__((ext_vector_type(2)));
typedef float        v8f   __attribute__((ext_vector_type(8)));

// Flat pointers to LDS carry the byte offset in bits [31:0] (aperture above),
// so truncation yields the wave-relative LDS address TDM/async ops need.
__device__ __forceinline__ unsigned lds_byte_off(const void* p) {
  return (unsigned)(uintptr_t)p;
}

__device__ __forceinline__ void wait_tensorcnt0() {
#if __has_builtin(__builtin_amdgcn_s_wait_tensorcnt)
  __builtin_amdgcn_s_wait_tensorcnt(0);
#else
  asm volatile("s_wait_tensorcnt 0x0" ::: "memory");
#endif
}

// ---------------------------------------------------------------------------
// Kernel 1: channel-max + horizontal sliding max (window 64, edge replicate).
// One 512-thread block per (batch, row). Van Herk via 64-block prefix/suffix
// Hillis-Steele max scans in LDS.
// ---------------------------------------------------------------------------
__global__ __launch_bounds__(WW) void hmax_kernel(const float* __restrict__ x,
                                                  float* __restrict__ hbuf) {
  __shared__ float pfx[WW];
  __shared__ float sfx[WW];
  const int t   = threadIdx.x;              // column 0..511
  const int rid = blockIdx.x;               // 0 .. BATCH*HH-1
  const int b   = rid >> 9;
  const int r   = rid & (HH - 1);

  // channel max (fold C=3 before pooling: max commutes with max-pool)
  const size_t rowbase = ((size_t)(b * CHN) * HH + r) * WW;
  float v = x[rowbase + t];
  v = fmaxf(v, x[rowbase + (size_t)HH * WW + t]);
  v = fmaxf(v, x[rowbase + 2 * (size_t)HH * WW + t]);

  pfx[t] = v;
  sfx[t] = v;
  __syncthreads();

  const int l = t & (KWIN - 1);
#pragma unroll
  for (int off = 1; off < KWIN; off <<= 1) {
    float pv = (l >= off)       ? pfx[t - off] : -__builtin_inff();
    float sv = (l + off < KWIN) ? sfx[t + off] : -__builtin_inff();
    __syncthreads();
    pfx[t] = fmaxf(pfx[t], pv);   // prefix-max within aligned 64-block
    sfx[t] = fmaxf(sfx[t], sv);   // suffix-max within aligned 64-block
    __syncthreads();
  }

  // window [t-31, t+32] clamped; van Herk combine with same-block edge cases
  const int lo = max(t - PL, 0);
  const int hi = min(t + PR, WW - 1);
  float o;
  if ((lo >> 6) == (hi >> 6))
    o = ((lo & 63) == 0) ? pfx[hi] : sfx[lo];
  else
    o = fmaxf(sfx[lo], pfx[hi]);

  hbuf[((size_t)b * HH + r) * WW + t] = o;
}

// ---------------------------------------------------------------------------
// Kernel 2: vertical sliding max (window 64, edge replicate) + partial sums.
// 64 threads per block; block owns a 64x64 output tile of one batch plane.
// The in-range row span (up to 127 rows x 64 cols, stride 512) is staged into
// LDS by ONE Tensor Data Mover descriptor (tensor_load_to_lds, TENSORcnt),
// issued from wave 0. Replicate padding is realized by clamping the staged
// row index at read time (edge rows equal their replicas bit-for-bit).
// ---------------------------------------------------------------------------
#define SROWS 127
#define TW    64

__global__ __launch_bounds__(TW) void vmax_sum_kernel(const float* __restrict__ hbuf,
                                                      float* __restrict__ partials) {
  __shared__ float dbuf[SROWS * TW];   // 32.5 KB staged input tile
  __shared__ float sfxv[TW * TW];      // 16 KB per-column suffix maxes
  __shared__ float red[TW];

  const int t  = threadIdx.x;          // local column 0..63
  const int id = blockIdx.x;           // 0..1023
  const int b  = id >> 6;
  const int rb = (id >> 3) & 7;        // output row block (x64)
  const int cb = id & 7;               // column block (x64)

  const int r0      = rb * 64;
  const int c0      = cb * 64;
  const int g_lo    = max(r0 - PL, 0);           // first in-range input row
  const int g_hi    = min(r0 + 95, HH - 1);      // last in-range input row
  const int s_start = g_lo - (r0 - PL);          // staged offset of g_lo
  const int cnt     = g_hi - g_lo + 1;           // rows actually loaded
  const int s_end   = s_start + cnt - 1;

  if (t < 32) {                        // wave 0 issues the TDM DMA once
    unsigned long long ga = (unsigned long long)(uintptr_t)
        (hbuf + ((size_t)b * HH + g_lo) * WW + c0);

    // D# group 0: count=1 | lds_addr | global_addr[56:0] | type=2 ("image")
    u32x4 g0;
    g0[0] = 1u;
    g0[1] = lds_byte_off(&dbuf[s_start * TW]);
    g0[2] = (unsigned)ga;
    g0[3] = ((unsigned)(ga >> 32) & 0x01FFFFFFu) | 0x80000000u;

    // D# group 1: data_size=4B; tensor 512 x cnt (stride 512); tile 64 x cnt
    u32x8 g1;
    g1[0] = 2u << 16;                            // data_size = 4 bytes
    g1[1] = ((unsigned)WW  & 0xFFFFu) << 16;     // tensor_dim0[15:0] = 512
    g1[2] = ((unsigned)cnt & 0xFFFFu) << 16;     // tensor_dim1[15:0] = cnt
    g1[3] = ((unsigned)TW) << 16;                // tile_dim0 = 64 elements
    g1[4] = (unsigned)cnt;                       // tile_dim1 = cnt rows
    g1[5] = (unsigned)WW;                        // tensor_dim0_stride = 512
    g1[6] = 0u;
    g1[7] = 0u;

    asm volatile("tensor_load_to_lds %0, %1"
                 :: "s"(g0), "s"(g1)
                 : "memory");
    wait_tensorcnt0();
  }
  __syncthreads();

  // Per-column (thread-private column => no barriers needed below):
  // suffix max of staged block [0..63]; lower clamp realizes top replication
  float run = -__builtin_inff();
  for (int i = 63; i >= 0; --i) {
    const int ci = i < s_start ? s_start : i;
    run = fmaxf(run, dbuf[ci * TW + t]);
    sfxv[i * TW + t] = run;
  }

  // output local r: max over staged [r, r+63] = max(S[r], P[r+63]);
  // upper clamp on the prefix stream realizes bottom replication
  float acc = sfxv[t];                           // r = 0 (exactly block 0)
  float p = -__builtin_inff();
  for (int r = 1; r < 64; ++r) {
    const int j  = r + 63;
    const int cj = j > s_end ? s_end : j;
    p = fmaxf(p, dbuf[cj * TW + t]);
    acc += fmaxf(sfxv[r * TW + t], p);
  }

  // deterministic fixed-order block reduction
  red[t] = acc;
  __syncthreads();
#pragma unroll
  for (int s = TW / 2; s > 0; s >>= 1) {
    if (t < s) red[t] += red[t + s];
    __syncthreads();
  }
  if (t == 0) partials[id] = red[0];
}

// ---------------------------------------------------------------------------
// Kernel 3: final reduction of 1024 partials via the matrix pipe.
// One wave; 16 chained V_WMMA_F32_16X16X4_F32 with a ones B-matrix:
// D = A*1 + C => every C column accumulates the row-sums, so the total of all
// streamed A elements = sum of column 0 = sum over lanes 0 and 16 of the 8
// C VGPRs (16x16 f32 C layout). Deterministic, f32 RNE accumulation.
// ---------------------------------------------------------------------------
__global__ __launch_bounds__(32) void finalize_kernel(const float* __restrict__ partials,
                                                      float* __restrict__ out) {
  const int t = threadIdx.x;   // 0..31, one full wave (EXEC all ones for WMMA)
  float tot;
#if __has_builtin(__builtin_amdgcn_wmma_f32_16x16x4_f32)
  v8f c = {};
  v2f ones;
  ones[0] = 1.0f;
  ones[1] = 1.0f;
#pragma unroll
  for (int i = 0; i < 16; ++i) {
    v2f a;
    a[0] = partials[i * 64 + t];        // any placement works: each C column sums all of A
    a[1] = partials[i * 64 + 32 + t];
    c = __builtin_amdgcn_wmma_f32_16x16x4_f32(false, a, false, ones,
                                              (short)0, c, false, false);
  }
  float s = c[0] + c[1] + c[2] + c[3] + c[4] + c[5] + c[6] + c[7];
  tot = s + __shfl(s, t ^ 16, 32);      // col 0: M=0..7 @ lane, M=8..15 @ lane^16
#else
  tot = 0.0f;
  if (t == 0)
    for (int i = 0; i < 1024; ++i) tot += partials[i];
#endif
  if (t == 0) out[0] = fabsf(1.0f - tot / 4194304.0f);   // BATCH*HH*WW
}

// ---------------------------------------------------------------------------
extern "C" void kernel_launch(void* const* d_in, const int* in_sizes, int n_in,
                              void* d_out, int out_size, void* d_ws, size_t ws_size,
                              hipStream_t stream) {
  (void)in_sizes; (void)n_in; (void)out_size; (void)ws_size;
  const float* x  = (const float*)d_in[0];
  float* out      = (float*)d_out;
  float* hbuf     = (float*)d_ws;          // 16.78 MB intermediate (lives in L2)
  float* partials = hbuf + NPIX;           // 1024 floats

  hmax_kernel<<<dim3(BATCH * HH), dim3(WW), 0, stream>>>(x, hbuf);
  vmax_sum_kernel<<<dim3(BATCH * 8 * 8), dim3(TW), 0, stream>>>(hbuf, partials);
  finalize_kernel<<<dim3(1), dim3(32), 0, stream>>>(partials, out);
}